// GINBlock_79147657331006
// MI455X (gfx1250) — compile-verified
//
#include <hip/hip_runtime.h>

typedef __attribute__((ext_vector_type(2))) float v2f;
typedef __attribute__((ext_vector_type(8))) float v8f;
typedef __attribute__((ext_vector_type(4))) unsigned int u32x4;
typedef __attribute__((ext_vector_type(8))) int i32x8;
typedef __attribute__((ext_vector_type(4))) int i32x4;

#define N_NODES  100000
#define N_EDGES  1600000
#define DF       128
#define SW       132   // padded LDS row stride for weights (banks: 132%64=4)
#define SY       132   // padded LDS row stride for per-wave x/y tile
#define BN_EPS   1e-3f

// ---------------- zero the aggregation buffer ----------------
__global__ void gin_zero_kernel(float* __restrict__ p, int n4) {
  int i = blockIdx.x * blockDim.x + threadIdx.x;
  if (i < n4) ((float4*)p)[i] = make_float4(0.f, 0.f, 0.f, 0.f);
}

// ---------------- edge scatter-add: agg[dst] += h[src] ----------------
// one wave per edge; lane covers 4 consecutive features (32*4 = 128)
__global__ void gin_scatter_kernel(const float* __restrict__ h,
                                   const long long* __restrict__ ei,
                                   float* __restrict__ agg) {
  long long t = (long long)blockIdx.x * blockDim.x + threadIdx.x;
  long long e = t >> 5;
  int lane = (int)(t & 31);
  if (e >= N_EDGES) return;
  long long src = ei[2 * e];
  long long dst = ei[2 * e + 1];
  const float4 v = *(const float4*)(h + src * DF + lane * 4);
  float* p = agg + dst * DF + lane * 4;
  unsafeAtomicAdd(p + 0, v.x);
  unsafeAtomicAdd(p + 1, v.y);
  unsafeAtomicAdd(p + 2, v.z);
  unsafeAtomicAdd(p + 3, v.w);
}

// Build a 2D TDM descriptor: 128x128 f32 tile, global row stride 128,
// LDS padded to stride 132 via pad_interval=128 DW / pad_amount=4 DW.
__device__ __forceinline__ void tdm_load_weight(const float* gsrc, float* lds_dst) {
  unsigned long long ga = (unsigned long long)(uintptr_t)gsrc;
  unsigned int la = (unsigned int)(uintptr_t)lds_dst;  // addr[31:0] = LDS offset

  u32x4 g0;
  g0[0] = 1u;                                    // count=1, user descriptor
  g0[1] = la;                                    // lds_addr
  g0[2] = (unsigned int)(ga & 0xFFFFFFFFu);      // global_addr[31:0]
  g0[3] = (unsigned int)((ga >> 32) & 0x01FFFFFFu) | (2u << 30);  // [56:32] | type=2

  i32x8 g1;
  g1[0] = (int)((2u << 16)     // data_size = 4 bytes
              | (1u << 20)     // pad_enable
              | (6u << 22)     // pad_interval: 128 DWORDs
              | (3u << 25));   // pad_amount:   4 DWORDs  -> stride 132
  g1[1] = (int)(128u << 16);   // abar_addr=0 | tensor_dim0 lo16 = 128
  g1[2] = (int)(128u << 16);   // tensor_dim0 hi16 = 0 | tensor_dim1 lo16 = 128
  g1[3] = (int)(128u << 16);   // tensor_dim1 hi16 = 0 | tile_dim0 = 128
  g1[4] = (int)128u;           // tile_dim1 = 128 | tile_dim2 = 0
  g1[5] = (int)128u;           // tensor_dim0_stride lo32 = 128
  g1[6] = 0;                   // stride0 hi | stride1 lo (unused, 2D)
  g1[7] = 0;

  i32x4 gz4 = (i32x4){0, 0, 0, 0};
  i32x8 gz8 = (i32x8){0, 0, 0, 0, 0, 0, 0, 0};
  __builtin_amdgcn_tensor_load_to_lds(g0, g1, gz4, gz4, gz8, 0);
}

// ---------------- fused (h+agg)@W1 + b1 -> BN -> ReLU -> @W2 + b2 -> ReLU ----
// block = 256 threads = 8 waves; each wave owns 16 node rows (block: 128 rows).
// f32 WMMA 16x16x4, K-loop over 128. W1/W2 arrive via TDM (W2 overlaps GEMM1).
__global__ __launch_bounds__(256) void gin_fused_kernel(
    const float* __restrict__ h,   const float* __restrict__ agg,
    const float* __restrict__ W1,  const float* __restrict__ b1,
    const float* __restrict__ gma, const float* __restrict__ bta,
    const float* __restrict__ mmean, const float* __restrict__ mvar,
    const float* __restrict__ W2,  const float* __restrict__ b2,
    float* __restrict__ out) {
  __shared__ float sW1[DF * SW];       // 67584 B
  __shared__ float sW2[DF * SW];       // 67584 B
  __shared__ float sY[8 * 16 * SY];    // 67584 B (per-wave 16x128 tiles)

  const int tid  = threadIdx.x;
  const int lane = tid & 31;
  const int wave = tid >> 5;
  const int hl   = lane >> 4;    // 0 or 1 (lane half)
  const int l16  = lane & 15;

  const int row0w = blockIdx.x * 128 + wave * 16;
  float* ytile = &sY[wave * 16 * SY];

  // Wave 0 kicks off both weight DMAs up front; W2 overlaps GEMM1.
  if (wave == 0) {
    tdm_load_weight(W1, sW1);
    tdm_load_weight(W2, sW2);
  }

  // fold b1 + BN into per-column scale/shift; fetch b2
  float scale[8], shift[8], bias2[8];
  #pragma unroll
  for (int ct = 0; ct < 8; ++ct) {
    int n = ct * 16 + l16;
    float iv = rsqrtf(mvar[n] + BN_EPS);
    float sc = iv * gma[n];
    scale[ct] = sc;
    shift[ct] = (b1[n] - mmean[n]) * sc + bta[n];
    bias2[ct] = b2[n];
  }

  // stage x = h + agg (16 rows per wave) -> ytile  (clamp tail rows)
  for (int idx = lane; idx < 16 * 32; idx += 32) {
    int r = idx >> 5, c4 = (idx & 31) * 4;
    int g = row0w + r; if (g > N_NODES - 1) g = N_NODES - 1;
    float4 a = *(const float4*)&h[(long long)g * DF + c4];
    float4 b = *(const float4*)&agg[(long long)g * DF + c4];
    a.x += b.x; a.y += b.y; a.z += b.z; a.w += b.w;
    *(float4*)&ytile[r * SY + c4] = a;
  }

  if (wave == 0) __builtin_amdgcn_s_wait_tensorcnt(1);  // W1 landed (in-order)
  __syncthreads();

  v8f acc[8];
  #pragma unroll
  for (int ct = 0; ct < 8; ++ct) acc[ct] = (v8f){0,0,0,0,0,0,0,0};

  // ---- GEMM1: (16x128) @ (128x128) via 16x16x4 f32 WMMA ----
  for (int k = 0; k < DF; k += 4) {
    // A frag: lane (hl,l16) holds row l16, K = k + 2*hl + {0,1}
    v2f a = *(const v2f*)&ytile[l16 * SY + k + 2 * hl];
    #pragma unroll
    for (int ct = 0; ct < 8; ++ct) {
      int n = ct * 16 + l16;
      v2f b;
      b.x = sW1[(k + 2 * hl    ) * SW + n];
      b.y = sW1[(k + 2 * hl + 1) * SW + n];
      acc[ct] = __builtin_amdgcn_wmma_f32_16x16x4_f32(
          false, a, false, b, (short)0, acc[ct], false, false);
    }
  }

  // ---- bias1 + BN + ReLU, write intermediate back to own ytile ----
  #pragma unroll
  for (int ct = 0; ct < 8; ++ct) {
    #pragma unroll
    for (int v = 0; v < 8; ++v) {
      float val = acc[ct][v] * scale[ct] + shift[ct];
      val = fmaxf(val, 0.f);
      ytile[(v + 8 * hl) * SY + ct * 16 + l16] = val;   // C layout: m = v+8*hl, n = l16
    }
  }

  if (wave == 0) __builtin_amdgcn_s_wait_tensorcnt(0);  // W2 landed
  __syncthreads();

  #pragma unroll
  for (int ct = 0; ct < 8; ++ct) acc[ct] = (v8f){0,0,0,0,0,0,0,0};

  // ---- GEMM2 ----
  for (int k = 0; k < DF; k += 4) {
    v2f a = *(const v2f*)&ytile[l16 * SY + k + 2 * hl];
    #pragma unroll
    for (int ct = 0; ct < 8; ++ct) {
      int n = ct * 16 + l16;
      v2f b;
      b.x = sW2[(k + 2 * hl    ) * SW + n];
      b.y = sW2[(k + 2 * hl + 1) * SW + n];
      acc[ct] = __builtin_amdgcn_wmma_f32_16x16x4_f32(
          false, a, false, b, (short)0, acc[ct], false, false);
    }
  }

  // ---- bias2 + ReLU + store ----
  #pragma unroll
  for (int ct = 0; ct < 8; ++ct) {
    #pragma unroll
    for (int v = 0; v < 8; ++v) {
      int g = row0w + v + 8 * hl;
      if (g < N_NODES) {
        float val = fmaxf(acc[ct][v] + bias2[ct], 0.f);
        out[(long long)g * DF + ct * 16 + l16] = val;
      }
    }
  }
}

extern "C" void kernel_launch(void* const* d_in, const int* in_sizes, int n_in,
                              void* d_out, int out_size, void* d_ws, size_t ws_size,
                              hipStream_t stream) {
  const float*     h     = (const float*)d_in[0];
  const long long* ei    = (const long long*)d_in[1];
  // d_in[2] = num_nodes scalar (compile-time constant here)
  const float*     W1    = (const float*)d_in[3];
  const float*     b1    = (const float*)d_in[4];
  const float*     gma   = (const float*)d_in[5];
  const float*     bta   = (const float*)d_in[6];
  const float*     mmean = (const float*)d_in[7];
  const float*     mvar  = (const float*)d_in[8];
  const float*     W2    = (const float*)d_in[9];
  const float*     b2    = (const float*)d_in[10];
  float* out = (float*)d_out;
  float* agg = (float*)d_ws;   // N_NODES*DF floats = 51.2 MB scratch

  int n4 = N_NODES * DF / 4;
  gin_zero_kernel<<<(n4 + 255) / 256, 256, 0, stream>>>(agg, n4);

  long long totalT = (long long)N_EDGES * 32;
  gin_scatter_kernel<<<(int)((totalT + 255) / 256), 256, 0, stream>>>(h, ei, agg);

  int nblk = (N_NODES + 127) / 128;   // 782
  gin_fused_kernel<<<nblk, 256, 0, stream>>>(h, agg, W1, b1, gma, bta,
                                             mmean, mvar, W2, b2, out);
}